// CDist_32633161515324
// MI455X (gfx1250) — compile-verified
//
#include <hip/hip_runtime.h>
#include <hip/hip_bf16.h>

typedef __attribute__((ext_vector_type(2))) float v2f;
typedef __attribute__((ext_vector_type(8))) float v8f;

// One wave32 computes one 16x16 tile of the 2048x2048 distance matrix.
// D = 64 (fixed by the reference).
//
// dist^2 = ||x||^2 + ||y||^2 - 2 x.y  computed entirely with
// V_WMMA_F32_16X16X4_F32:
//   - 16 K-steps accumulate  -2 * X * Y^T   (B fragment pre-scaled by -2)
//   - 1 augmented K-step adds xnorm_i + ynorm_j via A=[xn,1,0,0], B=[1,yn,0,0]
__global__ __launch_bounds__(256) void cdist_wmma_f32(
    const float* __restrict__ X, const float* __restrict__ Y,
    float* __restrict__ Out, int N, int M) {
  const int lane = threadIdx.x & 31;
  const int wave = threadIdx.x >> 5;

  const int tilesM = M >> 4;
  const int tilesN = N >> 4;
  const int tileId = blockIdx.x * 8 + wave;
  if (tileId >= tilesN * tilesM) return;   // wave-uniform; EXEC stays all-1s
  const int ti = tileId / tilesM;
  const int tj = tileId - ti * tilesM;

  const int l16 = lane & 15;
  const int hi  = lane >> 4;               // 0: K pair {0,1}; 1: K pair {2,3}

  // A fragment source: row of X.  B fragment source: row of Y (B = Y^T).
  const float* xr = X + (size_t)(ti * 16 + l16) * 64 + 2 * hi;
  const float* yr = Y + (size_t)(tj * 16 + l16) * 64 + 2 * hi;

  v8f acc = {};
  float axx = 0.0f, byy = 0.0f;

#pragma unroll
  for (int kk = 0; kk < 16; ++kk) {
    v2f a = *(const v2f*)(xr + 4 * kk);    // X[row][4kk+2hi .. +1]
    v2f b = *(const v2f*)(yr + 4 * kk);    // Y[col][4kk+2hi .. +1]
    axx += a.x * a.x + a.y * a.y;          // per-lane partial ||x_row||^2
    byy += b.x * b.x + b.y * b.y;          // per-lane partial ||y_col||^2
    v2f bneg;
    bneg.x = -2.0f * b.x;
    bneg.y = -2.0f * b.y;
    acc = __builtin_amdgcn_wmma_f32_16x16x4_f32(
        /*neg_a=*/false, a, /*neg_b=*/false, bneg,
        /*c_mod=*/(short)0, acc, /*reuse_a=*/false, /*reuse_b=*/false);
  }

  // Combine the two K-halves (lanes l and l+16 hold complementary pairs).
  float xn = axx + __shfl_xor(axx, 16, 32);   // ||x_{ti*16+l16}||^2
  float yn = byy + __shfl_xor(byy, 16, 32);   // ||y_{tj*16+l16}||^2

  // Augmented step: A[i][64..67] = [xn_i, 1, 0, 0]; B[64..67][j] = [1, yn_j, 0, 0]
  v2f aE, bE;
  if (hi == 0) {
    aE.x = xn;   aE.y = 1.0f;
    bE.x = 1.0f; bE.y = yn;
  } else {
    aE.x = 0.0f; aE.y = 0.0f;
    bE.x = 0.0f; bE.y = 0.0f;
  }
  acc = __builtin_amdgcn_wmma_f32_16x16x4_f32(
      false, aE, false, bE, (short)0, acc, false, false);

  // C/D layout: VGPR v, lanes 0-15 -> (M=v, N=lane); lanes 16-31 -> (M=v+8).
  const int orow0 = ti * 16 + hi * 8;
  const int ocol  = tj * 16 + l16;
  float* outp = Out + (size_t)orow0 * (size_t)M + ocol;
#pragma unroll
  for (int v = 0; v < 8; ++v) {
    float d = acc[v];
    d = d > 0.0f ? __builtin_sqrtf(d) : 0.0f;   // clamp tiny negatives
    outp[(size_t)v * (size_t)M] = d;
  }
}

extern "C" void kernel_launch(void* const* d_in, const int* in_sizes, int n_in,
                              void* d_out, int out_size, void* d_ws, size_t ws_size,
                              hipStream_t stream) {
  const float* X = (const float*)d_in[0];
  const float* Y = (const float*)d_in[1];
  float* Out = (float*)d_out;

  const int D = 64;
  const int N = in_sizes[0] / D;   // 2048
  const int M = in_sizes[1] / D;   // 2048

  const int tiles = (N >> 4) * (M >> 4);       // 16384 tiles of 16x16
  const int blocks = (tiles + 7) / 8;          // 8 wave32 tiles per block
  cdist_wmma_f32<<<blocks, 256, 0, stream>>>(X, Y, Out, N, M);
}